// SupportLowRankRNN_91036126806948
// MI455X (gfx1250) — compile-verified
//
#include <hip/hip_runtime.h>

#define H      2048
#define T_SEQ  1000
#define B_BATCH 32
#define I_IN   4
#define GDIM   8
#define ALPHA  0.2f
#define NSTD   0.05f
#define DEPTH  8          // async noise ring depth (steps in flight)

typedef __attribute__((ext_vector_type(2))) float v2f;
typedef __attribute__((ext_vector_type(8))) float v8f;

// -------- stacked parameter-row accessors (rows 0..15 of the WMMA A tile) ----
// rows 0-3: wi_weights (I,S,G)   rows 4-5: m_weights (R,S,G)
// rows 6-7: n_weights  (R,S,G)   rows 8-9: wo_weights (O,S,G)
// row  10 : h0_weights (S,G)     rows 11-15: zero pad
__device__ __forceinline__ float wrow(int row, int s, int g,
    const float* wi_w, const float* m_w, const float* n_w,
    const float* wo_w, const float* h0_w) {
  if (row < 4)   return wi_w[(row * 2 + s) * GDIM + g];
  if (row < 6)   return m_w[((row - 4) * 2 + s) * GDIM + g];
  if (row < 8)   return n_w[((row - 6) * 2 + s) * GDIM + g];
  if (row < 10)  return wo_w[((row - 8) * 2 + s) * GDIM + g];
  if (row == 10) return h0_w[s * GDIM + g];
  return 0.0f;
}
__device__ __forceinline__ float brow(int row, int s,
    const float* wi_b, const float* m_b, const float* n_b) {
  if (row < 4)  return wi_b[row * 2 + s];
  if (row < 6)  return m_b[(row - 4) * 2 + s];
  if (row < 8)  return n_b[(row - 6) * 2 + s];
  return 0.0f;   // wo and h0 have no bias
}

// ---------------------------------------------------------------------------
// Kernel 1: proxy-parameter precompute via V_WMMA_F32_16X16X4_F32.
// One wave per 16-wide H tile. A = stacked weight rows (16x8, two K=4 chunks),
// B = gaussian_basis chunk (8x16). D fragments masked by supports, biased,
// summed over supports, scattered to ws rows 0..10 (each H floats):
//   ws[0..3]=wi, ws[4..5]=m^T, ws[6..7]=n^T, ws[8..9]=wo^T, ws[10]=h0
// ---------------------------------------------------------------------------
__global__ __launch_bounds__(32) void proxy_params_wmma(
    const float* __restrict__ wi_w, const float* __restrict__ m_w,
    const float* __restrict__ n_w,  const float* __restrict__ wo_w,
    const float* __restrict__ wi_b, const float* __restrict__ m_b,
    const float* __restrict__ n_b,  const float* __restrict__ h0_w,
    const float* __restrict__ gb,   const float* __restrict__ sup,
    float* __restrict__ ws)
{
  const int lane  = threadIdx.x;
  const int hbase = blockIdx.x * 16;
  const int ncol  = lane & 15;          // D column / B column owned by lane
  const int kk    = (lane >> 4) * 2;    // k-pair base inside a K=4 chunk
  const int row   = lane & 15;          // A row owned by lane

  // B fragments: B[k][n] = gaussian_basis[g*H + h], chunk c covers g = c*4+{kk,kk+1}
  v2f b0, b1;
  b0.x = gb[(0 * 4 + kk + 0) * H + hbase + ncol];
  b0.y = gb[(0 * 4 + kk + 1) * H + hbase + ncol];
  b1.x = gb[(1 * 4 + kk + 0) * H + hbase + ncol];
  b1.y = gb[(1 * 4 + kk + 1) * H + hbase + ncol];

  float acc[8] = {0.f, 0.f, 0.f, 0.f, 0.f, 0.f, 0.f, 0.f};

  for (int s = 0; s < 2; ++s) {
    v2f a0, a1;
    a0.x = wrow(row, s, 0 * 4 + kk + 0, wi_w, m_w, n_w, wo_w, h0_w);
    a0.y = wrow(row, s, 0 * 4 + kk + 1, wi_w, m_w, n_w, wo_w, h0_w);
    a1.x = wrow(row, s, 1 * 4 + kk + 0, wi_w, m_w, n_w, wo_w, h0_w);
    a1.y = wrow(row, s, 1 * 4 + kk + 1, wi_w, m_w, n_w, wo_w, h0_w);

    v8f c = {};
    c = __builtin_amdgcn_wmma_f32_16x16x4_f32(false, a0, false, b0,
                                              (short)0, c, false, false);
    c = __builtin_amdgcn_wmma_f32_16x16x4_f32(false, a1, false, b1,
                                              (short)0, c, false, false);

    const float mask = sup[s * H + hbase + ncol];
#pragma unroll
    for (int r = 0; r < 8; ++r) {
      const int M = (lane < 16) ? r : (r + 8);
      acc[r] += (c[r] + brow(M, s, wi_b, m_b, n_b)) * mask;
    }
  }

#pragma unroll
  for (int r = 0; r < 8; ++r) {
    const int M = (lane < 16) ? r : (r + 8);
    if (M <= 10) ws[M * H + hbase + ncol] = acc[r];
  }
}

// ---------------------------------------------------------------------------
// Async copy of one thread's 32-byte noise slice into the LDS ring.
// Per ISA pseudocode, IOFFSET advances both the global and LDS addresses,
// so offset:16 issues the second half with identical operands.
// Tracked by ASYNCcnt (2 instructions per wave per step).
// ---------------------------------------------------------------------------
__device__ __forceinline__ void async_cp32B(unsigned lds, const float* g) {
  asm volatile("global_load_async_to_lds_b128 %0, %1, off"
               :: "v"(lds), "v"(g) : "memory");
  asm volatile("global_load_async_to_lds_b128 %0, %1, off offset:16"
               :: "v"(lds), "v"(g) : "memory");
}
__device__ __forceinline__ void wait_async_ring() {
  // oldest step's 2 loads complete when <= 2*(DEPTH-1)=14 remain outstanding
  asm volatile("s_wait_asynccnt 14" ::: "memory");
}

// ---------------------------------------------------------------------------
// Kernel 2: the sequential scan. One workgroup per batch (32 WGs x 256 thr,
// 8 waves). Each thread owns 8 contiguous hidden units; all params live in
// VGPRs. Per step: wave32 shfl_xor butterflies + 8-slot LDS combine for the
// rank-2 reductions (2 barriers/step). Noise is streamed via an 8-step-deep
// GLOBAL_LOAD_ASYNC_TO_LDS ring (64KB LDS); each thread copies and consumes
// its own region, so a same-wave s_wait_asynccnt suffices (no extra barrier).
// ---------------------------------------------------------------------------
__device__ __forceinline__ void ld8(float* d, const float* __restrict__ p) {
  float4 a = *(const float4*)p;
  float4 b = *(const float4*)(p + 4);
  d[0] = a.x; d[1] = a.y; d[2] = a.z; d[3] = a.w;
  d[4] = b.x; d[5] = b.y; d[6] = b.z; d[7] = b.w;
}

__global__ __launch_bounds__(256) void rnn_scan(
    const float* __restrict__ input,   // (B, T, I)
    const float* __restrict__ noise,   // (B, T, H)
    const float* __restrict__ ws,      // param rows 0..10, each H floats
    float* __restrict__ out)           // (B, T, 2)
{
  const int b    = blockIdx.x;
  const int tid  = threadIdx.x;
  const int lane = tid & 31;
  const int wv   = tid >> 5;
  const int hb   = tid * 8;            // this thread's hidden base index

  __shared__ float s_nz[DEPTH][H];     // 64 KB async noise ring
  __shared__ float s_in[T_SEQ * I_IN]; // 16 KB: whole input slice for batch b
  __shared__ float s_rk[8][2];         // per-wave kappa partials
  __shared__ float s_ro[8][2];         // per-wave output partials

  const float* inb = input + (size_t)b * T_SEQ * I_IN;
  for (int i = tid; i < T_SEQ * I_IN; i += 256) s_in[i] = inb[i];

  // prime the async noise ring for steps 0..DEPTH-1
  const float* nzb = noise + (size_t)b * T_SEQ * H;
  const unsigned nz_lds_base = (unsigned)(size_t)(&s_nz[0][hb]);
#pragma unroll
  for (int d = 0; d < DEPTH; ++d)
    async_cp32B(nz_lds_base + (unsigned)(d * H * 4), nzb + (size_t)d * H + hb);

  float wi0[8], wi1[8], wi2[8], wi3[8];
  float m0[8], m1[8], n0[8], n1[8], wo0[8], wo1[8], h[8], r[8];
  ld8(wi0, ws + 0 * H + hb);
  ld8(wi1, ws + 1 * H + hb);
  ld8(wi2, ws + 2 * H + hb);
  ld8(wi3, ws + 3 * H + hb);
  ld8(m0,  ws + 4 * H + hb);
  ld8(m1,  ws + 5 * H + hb);
  ld8(n0,  ws + 6 * H + hb);
  ld8(n1,  ws + 7 * H + hb);
  ld8(wo0, ws + 8 * H + hb);
  ld8(wo1, ws + 9 * H + hb);
  ld8(h,   ws + 10 * H + hb);          // h0
#pragma unroll
  for (int j = 0; j < 8; ++j) r[j] = tanhf(h[j]);

  __syncthreads();                     // s_in ready

  for (int t = 0; t < T_SEQ; ++t) {
    const int buf = t & (DEPTH - 1);

    // kappa_r = sum_h r[h] * n[h][r]  (uses r from previous step)
    float p0 = 0.f, p1 = 0.f;
#pragma unroll
    for (int j = 0; j < 8; ++j) {
      p0 = fmaf(r[j], n0[j], p0);
      p1 = fmaf(r[j], n1[j], p1);
    }
#pragma unroll
    for (int off = 16; off > 0; off >>= 1) {
      p0 += __shfl_xor(p0, off);
      p1 += __shfl_xor(p1, off);
    }
    if (lane == 0) { s_rk[wv][0] = p0; s_rk[wv][1] = p1; }

    // wait for this step's noise and pull this thread's slice from the ring
    wait_async_ring();
    float nz[8];
    {
      float4 na = *(const float4*)(&s_nz[buf][hb]);
      float4 nb4 = *(const float4*)(&s_nz[buf][hb + 4]);
      nz[0] = na.x; nz[1] = na.y; nz[2] = na.z; nz[3] = na.w;
      nz[4] = nb4.x; nz[5] = nb4.y; nz[6] = nb4.z; nz[7] = nb4.w;
    }

    __syncthreads();
    float k0 = 0.f, k1 = 0.f;
#pragma unroll
    for (int w2 = 0; w2 < 8; ++w2) { k0 += s_rk[w2][0]; k1 += s_rk[w2][1]; }

    const float xi0 = s_in[t * 4 + 0], xi1 = s_in[t * 4 + 1];
    const float xi2 = s_in[t * 4 + 2], xi3 = s_in[t * 4 + 3];

    // h += NSTD*noise + ALPHA*(-h + kappa@m^T + xin);  r = tanh(h);  q = r@wo
    float q0 = 0.f, q1 = 0.f;
#pragma unroll
    for (int j = 0; j < 8; ++j) {
      float xin = fmaf(xi3, wi3[j], fmaf(xi2, wi2[j], fmaf(xi1, wi1[j], xi0 * wi0[j])));
      float rec = fmaf(k1, m1[j], k0 * m0[j]);
      h[j] = fmaf(NSTD, nz[j], h[j]) + ALPHA * (rec + xin - h[j]);
      r[j] = tanhf(h[j]);
      q0 = fmaf(r[j], wo0[j], q0);
      q1 = fmaf(r[j], wo1[j], q1);
    }

    // refill the ring slot we just consumed (same thread region -> no hazard)
    if (t + DEPTH < T_SEQ)
      async_cp32B(nz_lds_base + (unsigned)(buf * H * 4),
                  nzb + (size_t)(t + DEPTH) * H + hb);

#pragma unroll
    for (int off = 16; off > 0; off >>= 1) {
      q0 += __shfl_xor(q0, off);
      q1 += __shfl_xor(q1, off);
    }
    if (lane == 0) { s_ro[wv][0] = q0; s_ro[wv][1] = q1; }
    __syncthreads();
    if (tid < 2) {
      float o = 0.f;
#pragma unroll
      for (int w2 = 0; w2 < 8; ++w2) o += s_ro[w2][tid];
      out[((size_t)b * T_SEQ + t) * 2 + tid] = o;
    }
  }
}

// ---------------------------------------------------------------------------
extern "C" void kernel_launch(void* const* d_in, const int* in_sizes, int n_in,
                              void* d_out, int out_size, void* d_ws, size_t ws_size,
                              hipStream_t stream) {
  (void)in_sizes; (void)n_in; (void)out_size; (void)ws_size;
  const float* input = (const float*)d_in[0];
  const float* noise = (const float*)d_in[1];
  const float* wi_w  = (const float*)d_in[2];
  const float* m_w   = (const float*)d_in[3];
  const float* n_w   = (const float*)d_in[4];
  const float* wo_w  = (const float*)d_in[5];
  const float* wi_b  = (const float*)d_in[6];
  const float* m_b   = (const float*)d_in[7];
  const float* n_b   = (const float*)d_in[8];
  const float* h0_w  = (const float*)d_in[9];
  const float* gb    = (const float*)d_in[10];
  const float* sup   = (const float*)d_in[11];
  float* ws  = (float*)d_ws;
  float* out = (float*)d_out;

  proxy_params_wmma<<<H / 16, 32, 0, stream>>>(wi_w, m_w, n_w, wo_w,
                                               wi_b, m_b, n_b, h0_w,
                                               gb, sup, ws);
  rnn_scan<<<B_BATCH, 256, 0, stream>>>(input, noise, ws, out);
}